// AGDN_87119116632167
// MI455X (gfx1250) — compile-verified
//
#include <hip/hip_runtime.h>
#include <hip/hip_bf16.h>

// AGDN (2-layer, heads=1, K=3) for gfx1250 / MI455X.
// Sizes fixed by the reference: N=50000, E=800000, D_IN=128, D_HID=D_OUT=64.

#define D_IN   128
#define D      64      // hidden == out
#define KHOPS  3
#define SLOPE  0.2f

typedef __attribute__((ext_vector_type(2))) float v2f;
typedef __attribute__((ext_vector_type(8))) float v8f;

// -----------------------------------------------------------------------------
// GEMM: C[N x 64] = A[N x Kdim] * W[64 x Kdim]^T   (all fp32, via WMMA f32 16x16x4)
// One wave computes one 16x16 tile of C. Block = 4 waves = the full 64-wide N dim.
// Grid.x = N/16 (N is a multiple of 16: 50000 = 3125*16).
// Fragment layouts per CDNA5 ISA 7.12.2:
//   A 16x4 f32 : lane (m = lane%16), VGPR0/1 = A[m][koff], A[m][koff+1], koff = 2*(lane/16)
//   B 4x16 f32 : lane (n = lane%16), VGPR0/1 = B[koff][n], B[koff+1][n]  (B = W^T => W[n][k])
//   C 16x16 f32: VGPR v = C[v + 8*(lane/16)][lane%16]
// -----------------------------------------------------------------------------
__global__ __launch_bounds__(128)
void agdn_gemm_wmma(const float* __restrict__ A,
                    const float* __restrict__ W,
                    float* __restrict__ C,
                    int Kdim)
{
    const int lane  = threadIdx.x & 31;
    const int wave  = threadIdx.x >> 5;          // 0..3 -> N-tile
    const int mrow  = (blockIdx.x << 4) + (lane & 15);   // global row of A
    const int ncol  = (wave << 4) + (lane & 15);         // global col of C (row of W)
    const int koff  = (lane >> 4) << 1;                  // 0 or 2

    const float* arow = A + (size_t)mrow * Kdim;
    const float* wrow = W + (size_t)ncol * Kdim;

    v8f acc = {};
    for (int k = 0; k < Kdim; k += 4) {
        v2f a = *(const v2f*)(arow + k + koff);
        v2f b = *(const v2f*)(wrow + k + koff);
        acc = __builtin_amdgcn_wmma_f32_16x16x4_f32(
            /*neg_a=*/false, a, /*neg_b=*/false, b,
            /*c_mod=*/(short)0, acc, /*reuse_a=*/false, /*reuse_b=*/false);
    }

    const int half = lane >> 4;                  // 0: rows 0..7, 1: rows 8..15
    float* crow = C + ((size_t)(blockIdx.x << 4) + 8 * half) * D + (wave << 4) + (lane & 15);
#pragma unroll
    for (int v = 0; v < 8; ++v)
        crow[(size_t)v * D] = acc[v];
}

// -----------------------------------------------------------------------------
// Scatter-add propagation: out[dst[e]][j] += in[src[e]][j]
// thread = (edge, feature). Wave-uniform edge id -> scalar src/dst loads; 32
// contiguous floats per wave on both gather and atomic sides.
// -----------------------------------------------------------------------------
__global__ __launch_bounds__(256)
void agdn_scatter_add(const float* __restrict__ in,
                      float* __restrict__ out,
                      const int* __restrict__ src,
                      const int* __restrict__ dst,
                      int E)
{
    unsigned tid = blockIdx.x * blockDim.x + threadIdx.x;
    unsigned total = (unsigned)E * D;
    if (tid >= total) return;
    int e = tid >> 6;          // D == 64
    int j = tid & 63;
    int s = src[e];
    int d = dst[e];
    atomicAdd(&out[(size_t)d * D + j], in[(size_t)s * D + j]);
}

// -----------------------------------------------------------------------------
// Hop attention + residual + bias (+ optional ELU). One wave32 per node; lane l
// owns features l and l+32. feats = 4 slabs of [N x 64] (h0, f1, f2, f3).
// score_k = leaky_relu( dot(att[0:64], h0) + dot(att[64:128], f_k) )
// out = sum_k softmax_k(score)*f_k + res + bias
// -----------------------------------------------------------------------------
__device__ __forceinline__ float wave_sum32(float v) {
#pragma unroll
    for (int off = 16; off > 0; off >>= 1)
        v += __shfl_xor(v, off, 32);
    return v;
}

__global__ __launch_bounds__(256)
void agdn_attn(const float* __restrict__ feats,   // 4 * N * 64
               const float* __restrict__ res,     // N * 64
               const float* __restrict__ att,     // 128
               const float* __restrict__ bias,    // 64
               float* __restrict__ out,           // N * 64
               int n, int applyElu)
{
    const int lane = threadIdx.x & 31;
    const int node = (blockIdx.x * blockDim.x + threadIdx.x) >> 5;
    if (node >= n) return;

    const float a0l = att[lane],      a0h = att[lane + 32];
    const float a1l = att[64 + lane], a1h = att[96 + lane];

    const size_t slab = (size_t)n * D;
    const float* f0 = feats + (size_t)node * D;

    // base = dot(att[0:64], h0)  (broadcast term of the concat)
    float h0l = f0[lane], h0h = f0[lane + 32];
    float base = wave_sum32(a0l * h0l + a0h * h0h);

    float flo[KHOPS + 1], fhi[KHOPS + 1], sc[KHOPS + 1];
#pragma unroll
    for (int k = 0; k <= KHOPS; ++k) {
        const float* fk = f0 + (size_t)k * slab;
        flo[k] = fk[lane];
        fhi[k] = fk[lane + 32];
        float s = base + wave_sum32(a1l * flo[k] + a1h * fhi[k]);
        sc[k] = (s > 0.0f) ? s : SLOPE * s;       // leaky_relu
    }

    // softmax over hops (all lanes hold identical sc[])
    float m = sc[0];
#pragma unroll
    for (int k = 1; k <= KHOPS; ++k) m = fmaxf(m, sc[k]);
    float esum = 0.0f, w[KHOPS + 1];
#pragma unroll
    for (int k = 0; k <= KHOPS; ++k) { w[k] = __expf(sc[k] - m); esum += w[k]; }
    float inv = 1.0f / esum;

    float ol = 0.0f, oh = 0.0f;
#pragma unroll
    for (int k = 0; k <= KHOPS; ++k) {
        float wk = w[k] * inv;
        ol += wk * flo[k];
        oh += wk * fhi[k];
    }

    const float* r = res + (size_t)node * D;
    ol += r[lane]      + bias[lane];
    oh += r[lane + 32] + bias[lane + 32];
    if (applyElu) {
        ol = (ol > 0.0f) ? ol : (__expf(ol) - 1.0f);
        oh = (oh > 0.0f) ? oh : (__expf(oh) - 1.0f);
    }
    out[(size_t)node * D + lane]      = ol;
    out[(size_t)node * D + lane + 32] = oh;
}

// -----------------------------------------------------------------------------
// Orchestration
// -----------------------------------------------------------------------------
extern "C" void kernel_launch(void* const* d_in, const int* in_sizes, int n_in,
                              void* d_out, int out_size, void* d_ws, size_t ws_size,
                              hipStream_t stream)
{
    const float* x     = (const float*)d_in[0];   // [N,128]
    const int*   eidx  = (const int*)  d_in[1];   // [2,E]
    const float* W1    = (const float*)d_in[2];   // [64,128]
    const float* att1  = (const float*)d_in[3];   // [128]
    const float* bias1 = (const float*)d_in[4];   // [64]
    const float* resW1 = (const float*)d_in[5];   // [64,128]
    const float* W2    = (const float*)d_in[6];   // [64,64]
    const float* att2  = (const float*)d_in[7];   // [128]
    const float* bias2 = (const float*)d_in[8];   // [64]

    const int N = in_sizes[0] / D_IN;             // 50000
    const int E = in_sizes[1] / 2;                // 800000
    const int* src = eidx;
    const int* dst = eidx + E;

    float* ws = (float*)d_ws;
    const size_t slab = (size_t)N * D;
    float* F   = ws;                 // 4 slabs: h0, f1, f2, f3
    float* RES = ws + 4 * slab;      // residual (layer 1)
    float* H   = ws + 5 * slab;      // layer-1 output / layer-2 input

    const dim3 gemmGrid(N / 16), gemmBlk(128);
    const dim3 scatGrid(((unsigned)E * D + 255) / 256), scatBlk(256);
    const dim3 attnGrid((N * 32 + 255) / 256), attnBlk(256);

    // ---------------- Layer 1 ----------------
    agdn_gemm_wmma<<<gemmGrid, gemmBlk, 0, stream>>>(x, W1, F, D_IN);        // h0
    agdn_gemm_wmma<<<gemmGrid, gemmBlk, 0, stream>>>(x, resW1, RES, D_IN);   // residual

    for (int k = 1; k <= KHOPS; ++k) {
        hipMemsetAsync(F + (size_t)k * slab, 0, slab * sizeof(float), stream);
        agdn_scatter_add<<<scatGrid, scatBlk, 0, stream>>>(
            F + (size_t)(k - 1) * slab, F + (size_t)k * slab, src, dst, E);
    }

    agdn_attn<<<attnGrid, attnBlk, 0, stream>>>(F, RES, att1, bias1, H, N, /*elu=*/1);

    // ---------------- Layer 2 ----------------
    agdn_gemm_wmma<<<gemmGrid, gemmBlk, 0, stream>>>(H, W2, F, D);           // h0 (also residual)

    for (int k = 1; k <= KHOPS; ++k) {
        hipMemsetAsync(F + (size_t)k * slab, 0, slab * sizeof(float), stream);
        agdn_scatter_add<<<scatGrid, scatBlk, 0, stream>>>(
            F + (size_t)(k - 1) * slab, F + (size_t)k * slab, src, dst, E);
    }

    agdn_attn<<<attnGrid, attnBlk, 0, stream>>>(F, /*res=*/F, att2, bias2,
                                                (float*)d_out, N, /*elu=*/0);
}